// LightGCNLayer_39745627357903
// MI455X (gfx1250) — compile-verified
//
#include <hip/hip_runtime.h>
#include <stdint.h>

#define TPB 256
#define FEAT 64
#define EPB 512  // edges staged per block (2 per thread via async b64)

// ---------- zero a u32 (or f32, same bit pattern) region ----------
__global__ void lg_zero_u32(uint32_t* __restrict__ p, int n) {
  int i = blockIdx.x * blockDim.x + threadIdx.x;
  int stride = gridDim.x * blockDim.x;
  for (; i < n; i += stride) p[i] = 0u;
}

// ---------- degree histograms ----------
__global__ void lg_degrees(const int* __restrict__ es, const int* __restrict__ ed,
                           uint32_t* __restrict__ deg_out, uint32_t* __restrict__ deg_in,
                           int E) {
  int i = blockIdx.x * blockDim.x + threadIdx.x;
  int stride = gridDim.x * blockDim.x;
  for (; i < E; i += stride) {
    atomicAdd(&deg_out[es[i]], 1u);
    atomicAdd(&deg_in[ed[i]], 1u);
  }
}

// ---------- norm_src = rsqrt(max(deg_out,1)) ----------
__global__ void lg_norms(const uint32_t* __restrict__ deg, float* __restrict__ nrm, int n) {
  int i = blockIdx.x * blockDim.x + threadIdx.x;
  int stride = gridDim.x * blockDim.x;
  for (; i < n; i += stride) {
    uint32_t d = deg[i];
    float fd = (float)(d > 1u ? d : 1u);
    nrm[i] = rsqrtf(fd);
  }
}

// ---------- gather -> scale -> scatter-add (the hot kernel) ----------
// One block handles 512 edges. Edge indices are staged into LDS with the
// CDNA5 async global->LDS data mover (ASYNCcnt path, b64 = 2 edges/lane),
// per-edge norms are then cached in LDS, and 16 lanes cooperate per edge:
// lane `sub` moves floats [sub*4, sub*4+4) of the 64-wide row via one
// global_load_b128 + 4 no-return global_atomic_add_f32.
__global__ __launch_bounds__(TPB) void lg_scatter(const float* __restrict__ feat,
                                                  const int* __restrict__ es,
                                                  const int* __restrict__ ed,
                                                  const float* __restrict__ nrm,
                                                  float* __restrict__ out, int E) {
  __shared__ uint32_t s_src[EPB];
  __shared__ uint32_t s_dst[EPB];
  __shared__ float    s_ns[EPB];

  const int tid  = threadIdx.x;
  const int base = blockIdx.x * EPB;
  const int e0   = base + 2 * tid;  // this thread stages edges e0, e0+1

  {
    uint32_t lds_s = (uint32_t)(uintptr_t)(&s_src[2 * tid]);
    uint32_t lds_d = (uint32_t)(uintptr_t)(&s_dst[2 * tid]);
    uint64_t g_s = (uint64_t)(uintptr_t)(es + e0);
    uint64_t g_d = (uint64_t)(uintptr_t)(ed + e0);
    if (e0 + 1 < E) {
      // CDNA5 async data mover: 8B Global -> LDS per lane, tracked by ASYNCcnt.
      asm volatile("global_load_async_to_lds_b64 %0, %1, off"
                   :: "v"(lds_s), "v"(g_s) : "memory");
      asm volatile("global_load_async_to_lds_b64 %0, %1, off"
                   :: "v"(lds_d), "v"(g_d) : "memory");
    } else if (e0 < E) {  // odd tail edge: 4B to avoid any OOB touch
      asm volatile("global_load_async_to_lds_b32 %0, %1, off"
                   :: "v"(lds_s), "v"(g_s) : "memory");
      asm volatile("global_load_async_to_lds_b32 %0, %1, off"
                   :: "v"(lds_d), "v"(g_d) : "memory");
    }
  }
  asm volatile("s_wait_asynccnt 0" ::: "memory");
  __syncthreads();

  // Cache per-edge source norms in LDS (one global broadcast per edge total,
  // instead of one per edge per 16-lane group in the hot loop).
  for (int j = tid; j < EPB; j += TPB) {
    s_ns[j] = (base + j < E) ? nrm[s_src[j]] : 0.0f;
  }
  __syncthreads();

  const int sub  = tid & 15;   // which float4 of the 64-wide row
  const int egrp = tid >> 4;   // which edge of the group of 16

  if (base + EPB <= E) {
    // Fast path: full chunk, no bounds checks.
#pragma unroll 4
    for (int iter = 0; iter < EPB / 16; ++iter) {
      const int elocal = iter * 16 + egrp;
      const uint32_t src = s_src[elocal];   // 16-lane LDS broadcast
      const uint32_t dst = s_dst[elocal];
      const float ns = s_ns[elocal];
      const float4 f =
          *(const float4*)(feat + (size_t)src * FEAT + sub * 4);  // b128 load
      float* o = out + (size_t)dst * FEAT + sub * 4;
      atomicAdd(o + 0, f.x * ns);           // global_atomic_add_f32 (no return)
      atomicAdd(o + 1, f.y * ns);
      atomicAdd(o + 2, f.z * ns);
      atomicAdd(o + 3, f.w * ns);
    }
  } else {
    // Tail chunk: per-edge guard.
    for (int iter = 0; iter < EPB / 16; ++iter) {
      const int elocal = iter * 16 + egrp;
      if (base + elocal < E) {
        const uint32_t src = s_src[elocal];
        const uint32_t dst = s_dst[elocal];
        const float ns = s_ns[elocal];
        const float4 f = *(const float4*)(feat + (size_t)src * FEAT + sub * 4);
        float* o = out + (size_t)dst * FEAT + sub * 4;
        atomicAdd(o + 0, f.x * ns);
        atomicAdd(o + 1, f.y * ns);
        atomicAdd(o + 2, f.z * ns);
        atomicAdd(o + 3, f.w * ns);
      }
    }
  }
}

// ---------- final scaling by rsqrt(max(deg_in,1)) ----------
__global__ void lg_scale(float* __restrict__ out, const uint32_t* __restrict__ deg_in,
                         int total) {
  int i = blockIdx.x * blockDim.x + threadIdx.x;
  int stride = gridDim.x * blockDim.x;
  for (; i < total; i += stride) {
    uint32_t d = deg_in[i >> 6];  // row = i / 64
    float fd = (float)(d > 1u ? d : 1u);
    out[i] *= rsqrtf(fd);
  }
}

extern "C" void kernel_launch(void* const* d_in, const int* in_sizes, int n_in,
                              void* d_out, int out_size, void* d_ws, size_t ws_size,
                              hipStream_t stream) {
  const float* feat = (const float*)d_in[0];
  const int*   es   = (const int*)d_in[1];
  const int*   ed   = (const int*)d_in[2];
  float*       out  = (float*)d_out;

  const int n_src = in_sizes[0] / FEAT;   // 100000
  const int E     = in_sizes[1];          // 1250000
  const int n_dst = out_size / FEAT;      // 50000 (robust vs scalar dtype)

  // workspace: deg_out[n_src] u32 | deg_in[n_dst] u32 | norm_src[n_src] f32
  uint32_t* deg_out = (uint32_t*)d_ws;
  uint32_t* deg_in  = deg_out + n_src;
  float*    nrm     = (float*)(deg_in + n_dst);

  // 1) zero degree counters and the output accumulator
  {
    int n = n_src + n_dst;  // contiguous region
    lg_zero_u32<<<(n + TPB - 1) / TPB, TPB, 0, stream>>>(deg_out, n);
    int blocks = (out_size + TPB - 1) / TPB;
    if (blocks > 4096) blocks = 4096;
    lg_zero_u32<<<blocks, TPB, 0, stream>>>((uint32_t*)out, out_size);
  }

  // 2) degree histograms
  {
    int blocks = (E + TPB - 1) / TPB;
    if (blocks > 2048) blocks = 2048;
    lg_degrees<<<blocks, TPB, 0, stream>>>(es, ed, deg_out, deg_in, E);
  }

  // 3) source norms
  lg_norms<<<(n_src + TPB - 1) / TPB, TPB, 0, stream>>>(deg_out, nrm, n_src);

  // 4) gather/scale/scatter (one block per 512-edge chunk)
  lg_scatter<<<(E + EPB - 1) / EPB, TPB, 0, stream>>>(feat, es, ed, nrm, out, E);

  // 5) destination norm scaling
  {
    int blocks = (out_size + TPB - 1) / TPB;
    if (blocks > 4096) blocks = 4096;
    lg_scale<<<blocks, TPB, 0, stream>>>(out, deg_in, out_size);
  }
}